// Net_7602092113944
// MI455X (gfx1250) — compile-verified
//
#include <hip/hip_runtime.h>
#include <hip/hip_bf16.h>
#include <math.h>

typedef __attribute__((ext_vector_type(2))) float v2f;
typedef __attribute__((ext_vector_type(8))) float v8f;

#define D_FEAT 16   // hidden/out width (both 16)

// ---------------------------------------------------------------------------
// GEMM: Y[nrows,16] = X[nrows,K] @ W[K,16] using V_WMMA_F32_16X16X4_F32.
// One wave owns a 16x16 output tile; K advances 4 per WMMA.
// W is pre-swizzled into LDS so the B fragment is one ds_load_b64 per step:
//   step s, lane l:  b = { W[k][n], W[k+1][n] },  k = 4*s + (l>=16 ? 2 : 0),
//                    n = l & 15          (B layout: V0->K{0,2}, V1->K{1,3})
// A fragment: lane l loads { X[row][k], X[row][k+1] } (same k rule, row = l&15).
// ---------------------------------------------------------------------------
template <int K>
__global__ __launch_bounds__(256) void gemm_wmma_f32(
    const float* __restrict__ X, const float* __restrict__ W,
    float* __restrict__ Y, int nrows)
{
    constexpr int STEPS = K / 4;
    __shared__ float bfrag[STEPS * 64];   // STEPS * 32 lanes * 2 floats

    const int tid = threadIdx.x;

    // Cooperative swizzle of W[K,16] into fragment-ready LDS layout.
    for (int i = tid; i < STEPS * 32; i += 256) {
        int step = i >> 5;
        int lane = i & 31;
        int n = lane & 15;
        int k = step * 4 + ((lane & 16) ? 2 : 0);
        bfrag[i * 2 + 0] = W[k * D_FEAT + n];
        bfrag[i * 2 + 1] = W[(k + 1) * D_FEAT + n];
    }
    __syncthreads();

    const int wave = tid >> 5;                  // 8 waves per block
    const int lane = tid & 31;
    const int tile = blockIdx.x * 8 + wave;     // 16-row output tile
    const int row0 = tile * 16;
    if (row0 >= nrows) return;                  // wave-uniform (EXEC stays all-1)

    const int m    = lane & 15;
    const int koff = (lane & 16) ? 2 : 0;
    const float* xrow = X + (size_t)(row0 + m) * K + koff;

    v8f acc = {};
#pragma unroll 4
    for (int s = 0; s < STEPS; ++s) {
        v2f a = *(const v2f*)(xrow + s * 4);                 // global_load_b64
        v2f b = *(const v2f*)(&bfrag[(s * 32 + lane) * 2]);  // ds_load_b64
        acc = __builtin_amdgcn_wmma_f32_16x16x4_f32(
            /*neg_a=*/false, a, /*neg_b=*/false, b,
            /*c_mod=*/(short)0, acc, /*reuse_a=*/false, /*reuse_b=*/false);
    }

    // D layout: VGPR r -> M = r (lanes 0-15) / 8+r (lanes 16-31), N = lane&15.
    const int mbase = (lane & 16) ? 8 : 0;
    const int n = lane & 15;
#pragma unroll
    for (int r = 0; r < 8; ++r)
        Y[(size_t)(row0 + mbase + r) * D_FEAT + n] = acc[r];
}

// ---------------------------------------------------------------------------
// Weighted scatter-add over edges: agg[dst] += msg[src] * w  (16 feats/edge).
// agg (6.4 MB) is L2-resident, so the atomics stay on-chip.
// ---------------------------------------------------------------------------
__global__ __launch_bounds__(256) void scatter_add_kernel(
    const float* __restrict__ feat, const long long* __restrict__ ei,
    const float* __restrict__ ew, float* __restrict__ agg, int nedges)
{
    int e = blockIdx.x * blockDim.x + threadIdx.x;
    if (e >= nedges) return;
    long long src = ei[e];
    long long dst = ei[(size_t)nedges + e];
    float w = ew[e];
    const float4* hp = (const float4*)(feat + (size_t)src * D_FEAT);
    float* ap = agg + (size_t)dst * D_FEAT;
#pragma unroll
    for (int i = 0; i < 4; ++i) {
        float4 v = hp[i];
        atomicAdd(ap + i * 4 + 0, v.x * w);
        atomicAdd(ap + i * 4 + 1, v.y * w);
        atomicAdd(ap + i * 4 + 2, v.z * w);
        atomicAdd(ap + i * 4 + 3, v.w * w);
    }
}

__global__ __launch_bounds__(256) void zero_kernel(float* __restrict__ p, int n)
{
    int i = blockIdx.x * blockDim.x + threadIdx.x;
    if (i < n) p[i] = 0.0f;
}

__global__ __launch_bounds__(256) void bias_relu_kernel(
    const float* __restrict__ agg, const float* __restrict__ b,
    float* __restrict__ h, int total)
{
    int i = blockIdx.x * blockDim.x + threadIdx.x;
    if (i < total) {
        float v = agg[i] + b[i & (D_FEAT - 1)];
        h[i] = v > 0.0f ? v : 0.0f;
    }
}

__global__ __launch_bounds__(256) void bias_logsoftmax_kernel(
    const float* __restrict__ agg, const float* __restrict__ b,
    float* __restrict__ out, int nnodes)
{
    int nd = blockIdx.x * blockDim.x + threadIdx.x;
    if (nd >= nnodes) return;
    float v[D_FEAT];
    float m = -INFINITY;
#pragma unroll
    for (int i = 0; i < D_FEAT; ++i) {
        v[i] = agg[(size_t)nd * D_FEAT + i] + b[i];
        m = fmaxf(m, v[i]);
    }
    float s = 0.0f;
#pragma unroll
    for (int i = 0; i < D_FEAT; ++i) s += __expf(v[i] - m);
    float lse = __logf(s) + m;   // log_softmax = v - logsumexp(v)
#pragma unroll
    for (int i = 0; i < D_FEAT; ++i)
        out[(size_t)nd * D_FEAT + i] = v[i] - lse;
}

extern "C" void kernel_launch(void* const* d_in, const int* in_sizes, int n_in,
                              void* d_out, int out_size, void* d_ws, size_t ws_size,
                              hipStream_t stream) {
    const float*     x  = (const float*)d_in[0];       // [N, 512]
    const long long* ei = (const long long*)d_in[1];   // [2, E] int64
    const float*     ew = (const float*)d_in[2];       // [E]
    const float*     W1 = (const float*)d_in[3];       // [512, 16]
    const float*     b1 = (const float*)d_in[4];       // [16]
    const float*     W2 = (const float*)d_in[5];       // [16, 16]
    const float*     b2 = (const float*)d_in[6];       // [16]
    float*           out = (float*)d_out;              // [N, 16]

    const int D_IN   = 512;
    const int nnodes = in_sizes[0] / D_IN;             // 100000
    const int nedges = in_sizes[2];                    // 6400000
    const int total  = nnodes * D_FEAT;                // 1.6M

    // Workspace: 3 x N*16 float buffers (19.2 MB total).
    float* bufA = (float*)d_ws;            // linear-transform output (both layers)
    float* bufB = bufA + (size_t)total;    // scatter accumulator (both layers)
    float* bufC = bufB + (size_t)total;    // relu(h) for layer 2
    (void)ws_size; (void)n_in; (void)out_size;

    const int tiles    = (nnodes + 15) / 16;
    const int gemmBlks = (tiles + 7) / 8;
    const int eltBlks  = (total + 255) / 256;
    const int edgeBlks = (nedges + 255) / 256;
    const int nodeBlks = (nnodes + 255) / 256;

    // ----- Layer 1: relu(scatter(x @ W1) + b1) -----
    zero_kernel<<<eltBlks, 256, 0, stream>>>(bufB, total);
    gemm_wmma_f32<512><<<gemmBlks, 256, 0, stream>>>(x, W1, bufA, nnodes);
    scatter_add_kernel<<<edgeBlks, 256, 0, stream>>>(bufA, ei, ew, bufB, nedges);
    bias_relu_kernel<<<eltBlks, 256, 0, stream>>>(bufB, b1, bufC, total);

    // ----- Layer 2: log_softmax(scatter(h @ W2) + b2) -----
    zero_kernel<<<eltBlks, 256, 0, stream>>>(bufB, total);
    gemm_wmma_f32<16><<<gemmBlks, 256, 0, stream>>>(bufC, W2, bufA, nnodes);
    scatter_add_kernel<<<edgeBlks, 256, 0, stream>>>(bufA, ei, ew, bufB, nedges);
    bias_logsoftmax_kernel<<<nodeBlks, 256, 0, stream>>>(bufB, b2, out, nnodes);
}